// ComputeEndLoss_12506944766668
// MI455X (gfx1250) — compile-verified
//
#include <hip/hip_runtime.h>
#include <stdint.h>

// Problem constants (reference: B=8, S=4096, N=4096, C=3, radius=0.1, nsample=16)
#define BATCH   8
#define SPTS    4096
#define NPTS    4096
#define NS      16
#define THREADS 256

typedef float v2f __attribute__((ext_vector_type(2)));
typedef float v8f __attribute__((ext_vector_type(8)));

__global__ void ComputeEndLoss_zero_kernel(float* out) { out[0] = 0.0f; }

// select one of 8 wave-uniform words by per-lane index k (3-level cndmask tree)
__device__ __forceinline__ unsigned sel8(unsigned b0, unsigned b1, unsigned b2, unsigned b3,
                                         unsigned b4, unsigned b5, unsigned b6, unsigned b7,
                                         int k) {
    unsigned t01 = (k & 1) ? b1 : b0;
    unsigned t23 = (k & 1) ? b3 : b2;
    unsigned t45 = (k & 1) ? b5 : b4;
    unsigned t67 = (k & 1) ? b7 : b6;
    unsigned t03 = (k & 2) ? t23 : t01;
    unsigned t47 = (k & 2) ? t67 : t45;
    return (k & 4) ? t47 : t03;
}

__global__ __launch_bounds__(THREADS)
void ComputeEndLoss_wmma_kernel(const float* __restrict__ recon,
                                const float* __restrict__ gt,
                                float* __restrict__ out)
{
    // (x, y, z, |p|^2) per gt point: 4096 * 16B = 64 KB LDS
    __shared__ float4 pts[NPTS];

    const int blocksPerBatch = SPTS / THREADS;
    const int b    = blockIdx.x / blocksPerBatch;
    const int s    = (blockIdx.x % blocksPerBatch) * THREADS + threadIdx.x;
    const int lane = threadIdx.x & 31;
    const bool lo  = lane < 16;

    const float* gbase = gt + (size_t)b * NPTS * 3;

    // ---- Stage 1: async global -> LDS copy (12B records into 16B slots;
    //      CDNA5 applies inst offset to BOTH global and LDS addresses) -------
    for (int p = threadIdx.x; p < NPTS; p += THREADS) {
        unsigned lds = (unsigned)(uintptr_t)(&pts[p]);
        uint64_t ga  = (uint64_t)(uintptr_t)(gbase + p * 3);
        asm volatile("global_load_async_to_lds_b32 %0, %1, off"          :: "v"(lds), "v"(ga) : "memory");
        asm volatile("global_load_async_to_lds_b32 %0, %1, off offset:4" :: "v"(lds), "v"(ga) : "memory");
        asm volatile("global_load_async_to_lds_b32 %0, %1, off offset:8" :: "v"(lds), "v"(ga) : "memory");
    }
    asm volatile("s_wait_asynccnt 0x0" ::: "memory");
    __syncthreads();

    // ---- Stage 2: |p|^2 into .w ------------------------------------------
    for (int p = threadIdx.x; p < NPTS; p += THREADS) {
        float4 v = pts[p];
        pts[p].w = fmaf(v.x, v.x, fmaf(v.y, v.y, v.z * v.z));
    }
    __syncthreads();

    // ---- Own query + A-matrix construction (once per wave) ----------------
    const float* q = recon + ((size_t)b * SPTS + s) * 3;
    const float qx = q[0], qy = q[1], qz = q[2];
    const float n1 = fmaf(qx, qx, fmaf(qy, qy, qz * qz));

    // A 16x4 f32 VGPR layout: VGPR0 = K0 (lanes 0-15) / K2 (lanes 16-31);
    //                         VGPR1 = K1 (lanes 0-15) / K3 (lanes 16-31).
    // A row_m = [x, y, z, n1]; tile LO = wave rows 0-15, tile HI = rows 16-31.
    const float zlw = __shfl(qz, lane & 15, 32);   // z of row (lane-16)
    const float nlw = __shfl(n1, lane & 15, 32);   // n1 of row (lane-16)
    const float xhi = __shfl(qx, lane | 16, 32);   // x of row (lane+16)
    const float yhi = __shfl(qy, lane | 16, 32);   // y of row (lane+16)
    v2f a_lo, a_hi;
    a_lo[0] = lo ? qx  : zlw;   a_lo[1] = lo ? qy  : nlw;
    a_hi[0] = lo ? xhi : qz;    a_hi[1] = lo ? yhi : n1;

    const float R2 = (float)(0.1 * 0.1);   // match reference fp32 threshold

    // k==0 fallback: all 16 slots become point N-1
    float4 last = pts[NPTS - 1];
    float fx = last.x, fy = last.y, fz = last.z;
    float sx = 0.f, sy = 0.f, sz = 0.f;
    int   cnt = 0;

    // ---- Main loop: 16 candidates per chunk, sq via WMMA, hits via ballot --
    for (int n0 = 0; n0 < NPTS; n0 += 16) {
        if ((n0 & 63) == 0) {                       // uniform early exit
            if (__ballot(cnt < NS) == 0ull) break;
        }
        // B 4x16 f32 (assumed layout: VGPR0 = K0 lo-lanes / K2 hi-lanes;
        //                             VGPR1 = K1 lo-lanes / K3 hi-lanes)
        // B col_n = [-2px, -2py, -2pz, 1]
        float4 c4 = pts[n0 + (lane & 15)];
        v2f bm;
        bm[0] = lo ? (-2.0f * c4.x) : (-2.0f * c4.z);
        bm[1] = lo ? (-2.0f * c4.y) : 1.0f;
        // C[m][n] = |p_n|^2 : per-lane broadcast in C/D layout
        const float w = c4.w;
        v8f cm = {w, w, w, w, w, w, w, w};

        // D[m][n] = n1_m + nrm_n - 2 q_m . p_n  (exact reference sq)
        v8f dA = __builtin_amdgcn_wmma_f32_16x16x4_f32(false, a_lo, false, bm,
                                                       (short)0, cm, false, false);
        v8f dB = __builtin_amdgcn_wmma_f32_16x16x4_f32(false, a_hi, false, bm,
                                                       (short)0, cm, false, false);

        // One ballot per D register = 32 hit bits (rows v & v+8, cols 0-15)
        unsigned bA0 = (unsigned)__ballot(dA[0] <= R2);
        unsigned bA1 = (unsigned)__ballot(dA[1] <= R2);
        unsigned bA2 = (unsigned)__ballot(dA[2] <= R2);
        unsigned bA3 = (unsigned)__ballot(dA[3] <= R2);
        unsigned bA4 = (unsigned)__ballot(dA[4] <= R2);
        unsigned bA5 = (unsigned)__ballot(dA[5] <= R2);
        unsigned bA6 = (unsigned)__ballot(dA[6] <= R2);
        unsigned bA7 = (unsigned)__ballot(dA[7] <= R2);
        unsigned bB0 = (unsigned)__ballot(dB[0] <= R2);
        unsigned bB1 = (unsigned)__ballot(dB[1] <= R2);
        unsigned bB2 = (unsigned)__ballot(dB[2] <= R2);
        unsigned bB3 = (unsigned)__ballot(dB[3] <= R2);
        unsigned bB4 = (unsigned)__ballot(dB[4] <= R2);
        unsigned bB5 = (unsigned)__ballot(dB[5] <= R2);
        unsigned bB6 = (unsigned)__ballot(dB[6] <= R2);
        unsigned bB7 = (unsigned)__ballot(dB[7] <= R2);

        // Route: lane m's row lives in ballot (m&7); hi half-row if (m&8);
        // tile picked by (m&16).
        unsigned tA = sel8(bA0, bA1, bA2, bA3, bA4, bA5, bA6, bA7, lane & 7);
        unsigned tB = sel8(bB0, bB1, bB2, bB3, bB4, bB5, bB6, bB7, lane & 7);
        unsigned t  = (lane & 16) ? tB : tA;
        t = (lane & 8) ? (t >> 16) : t;
        unsigned mask = t & 0xffffu;

        // Sparse accumulation over genuine hits, ascending candidate order
        while (mask && cnt < NS) {
            int j = __builtin_ctz(mask);
            mask &= mask - 1;
            float4 c = pts[n0 + j];
            if (cnt == 0) { fx = c.x; fy = c.y; fz = c.z; }
            sx += c.x; sy += c.y; sz += c.z;
            cnt++;
        }
    }

    // s = sum(hits) + (16-cnt)*first - 16*p1 ; dist = ||s||
    const float rem = (float)(NS - cnt);
    const float vx = fmaf(rem, fx, sx) - (float)NS * qx;
    const float vy = fmaf(rem, fy, sy) - (float)NS * qy;
    const float vz = fmaf(rem, fz, sz) - (float)NS * qz;
    float dist = sqrtf(fmaf(vx, vx, fmaf(vy, vy, vz * vz)));

    // output = (sum over B*S of dist) * 24 / (B*S*S)
    const float scale = (float)(24.0 / ((double)BATCH * (double)SPTS * (double)SPTS));
    float val = dist * scale;

    #pragma unroll
    for (int off = 16; off > 0; off >>= 1)
        val += __shfl_xor(val, off, 32);
    if (lane == 0)
        atomicAdd(out, val);
}

extern "C" void kernel_launch(void* const* d_in, const int* in_sizes, int n_in,
                              void* d_out, int out_size, void* d_ws, size_t ws_size,
                              hipStream_t stream) {
    (void)in_sizes; (void)n_in; (void)out_size; (void)d_ws; (void)ws_size;
    const float* recon = (const float*)d_in[0];   // (8, 4096, 3) fp32
    const float* gt    = (const float*)d_in[1];   // (8, 4096, 3) fp32
    float* out = (float*)d_out;                   // single fp32

    ComputeEndLoss_zero_kernel<<<1, 1, 0, stream>>>(out);

    const int blocks = BATCH * (SPTS / THREADS);  // 128 blocks x 256 threads
    ComputeEndLoss_wmma_kernel<<<blocks, THREADS, 0, stream>>>(recon, gt, out);
}